// ExpertEngine_22651657519439
// MI455X (gfx1250) — compile-verified
//
#include <hip/hip_runtime.h>
#include <hip/hip_bf16.h>

// ---------------- problem constants ----------------
#define NTOK   65536      // B*T tokens
#define CDIM   256        // n_embd
#define NEXP   64         // experts
#define DDIM   128        // expert dim
#define KSEL   4096       // tokens per expert (NTOK/16)

// LDS strides (floats) chosen to avoid 64-bank conflicts.
// These are reproduced by TDM LDS padding:
//   257 = 256 dwords + pad 1 dword   (pad_interval code 7, pad_amount code 0)
//   132 = 128 dwords + pad 4 dwords  (pad_interval code 6, pad_amount code 3)
#define XS_STRIDE 257
#define WS_STRIDE 257
#define HS_STRIDE 132

typedef float v2f __attribute__((ext_vector_type(2)));
typedef float v8f __attribute__((ext_vector_type(8)));
typedef unsigned int u32x4 __attribute__((ext_vector_type(4)));
typedef int i32x4 __attribute__((ext_vector_type(4)));
typedef int i32x8 __attribute__((ext_vector_type(8)));

// fp32 WMMA: D(16x16) = A(16x4) * B(4x16) + C   (exact fp32, matrix pipe)
__device__ __forceinline__ v8f wmma4(v2f a, v2f b, v8f c) {
  return __builtin_amdgcn_wmma_f32_16x16x4_f32(
      /*neg_a=*/false, a, /*neg_b=*/false, b,
      /*c_mod=*/(short)0, c, /*reuse_a=*/false, /*reuse_b=*/false);
}

// monotone float -> uint key (larger float => larger key)
__device__ __forceinline__ unsigned fkey(float f) {
  unsigned b = __float_as_uint(f);
  return (b & 0x80000000u) ? ~b : (b | 0x80000000u);
}

// ---------------------------------------------------------------------
// Tensor Data Mover helpers (CDNA5 TDM, CDNA5 ISA)
// ---------------------------------------------------------------------
__device__ __forceinline__ void tdm_load(u32x4 g0, i32x8 g1, i32x4 g2, i32x4 g3) {
#if __clang_major__ >= 23
  i32x8 g4 = {0, 0, 0, 0, 0, 0, 0, 0};
  __builtin_amdgcn_tensor_load_to_lds(g0, g1, g2, g3, g4, 0);
#else
  __builtin_amdgcn_tensor_load_to_lds(g0, g1, g2, g3, 0);
#endif
}

// D# group 0: count=1 | gather_mode | lds_addr | global_addr[56:0] | type=2
__device__ __forceinline__ u32x4 tdm_g0(unsigned lds_off, const void* gaddr,
                                        bool gather16) {
  unsigned long long ga = (unsigned long long)(uintptr_t)gaddr;
  u32x4 g;
  g.x = 1u | (gather16 ? (1u << 31) : 0u);   // count=1; gather_mode; 16-bit idx
  g.y = lds_off;                             // lds_addr (bytes)
  g.z = (unsigned)ga;                        // global_addr[31:0]
  g.w = (unsigned)((ga >> 32) & 0x01FFFFFFu) | (2u << 30);  // [56:32] | type=2
  return g;
}

// D# group 1: data_size | pad ctl | tensor_dim0/1 | tile_dim0/1 | dim0 stride
__device__ __forceinline__ i32x8 tdm_g1(unsigned dsz_code, unsigned pad_ivl,
                                        unsigned pad_amt, unsigned td0,
                                        unsigned td1, unsigned tl0,
                                        unsigned tl1, unsigned long long s0) {
  i32x8 g;
  g[0] = (int)((dsz_code << 16) | (1u << 20) |           // pad_enable
               (pad_ivl << 22) | (pad_amt << 25));
  g[1] = (int)((td0 & 0xFFFFu) << 16);                   // tensor_dim0[15:0]
  g[2] = (int)(((td0 >> 16) & 0xFFFFu) | ((td1 & 0xFFFFu) << 16));
  g[3] = (int)(((td1 >> 16) & 0xFFFFu) | ((tl0 & 0xFFFFu) << 16));
  g[4] = (int)(tl1 & 0xFFFFu);                           // tile_dim2 = 0
  g[5] = (int)(s0 & 0xFFFFFFFFull);                      // dim0_stride[31:0]
  g[6] = (int)((s0 >> 32) & 0xFFFFu);                    // dim1_stride = 0
  g[7] = 0;
  return g;
}

// =====================================================================
// Kernel 1: router GEMM via WMMA; writes logits transposed (E x NTOK)
//   block = 128 threads (4 waves), 64 tokens per block
// =====================================================================
__global__ __launch_bounds__(128)
void router_kernel(const float* __restrict__ x, const float* __restrict__ rw,
                   float* __restrict__ logitsT) {
  extern __shared__ char smemraw[];
  float* Xs = (float*)smemraw;                 // 64 x XS_STRIDE
  float* Ws = Xs + 64 * XS_STRIDE;             // 16 x WS_STRIDE
  const int tid  = threadIdx.x;
  const int tok0 = blockIdx.x * 64;

  const float4* xf4 = (const float4*)x;
  for (int i = tid; i < 64 * 64; i += 128) {   // 64 rows x 64 float4
    int r = i >> 6, c4 = i & 63;
    float4 v = xf4[(size_t)(tok0 + r) * 64 + c4];
    float* dst = Xs + r * XS_STRIDE + c4 * 4;
    dst[0] = v.x; dst[1] = v.y; dst[2] = v.z; dst[3] = v.w;
  }
  __syncthreads();

  const int wave = tid >> 5, lane = tid & 31;
  const int ml = lane & 15;            // M (A) / N (B,C,D) within half-wave
  const int ko = (lane >> 4) << 1;     // K offset: 0 for lanes 0-15, 2 for 16-31
  const int m0 = wave * 16;

  for (int nt = 0; nt < NEXP / 16; ++nt) {
    const float4* wf4 = (const float4*)(rw + (size_t)(nt * 16) * CDIM);
    for (int i = tid; i < 16 * 64; i += 128) {
      int r = i >> 6, c4 = i & 63;
      float4 v = wf4[r * 64 + c4];
      float* dst = Ws + r * WS_STRIDE + c4 * 4;
      dst[0] = v.x; dst[1] = v.y; dst[2] = v.z; dst[3] = v.w;
    }
    __syncthreads();

    v8f acc = {0.f, 0.f, 0.f, 0.f, 0.f, 0.f, 0.f, 0.f};
    const float* arow = Xs + (m0 + ml) * XS_STRIDE;
    const float* brow = Ws + ml * WS_STRIDE;     // B[k][n] = rw[n][k]
    #pragma unroll 8
    for (int k0 = 0; k0 < CDIM; k0 += 4) {
      v2f a; a.x = arow[k0 + ko]; a.y = arow[k0 + ko + 1];
      v2f b; b.x = brow[k0 + ko]; b.y = brow[k0 + ko + 1];
      acc = wmma4(a, b, acc);
    }
    const int eo = nt * 16 + ml;
    #pragma unroll
    for (int r = 0; r < 8; ++r) {
      int m = m0 + r + ((lane >> 4) << 3);       // C/D layout: M = r (+8 hi half)
      logitsT[(size_t)eo * NTOK + tok0 + m] = acc[r];
    }
    __syncthreads();
  }
}

// =====================================================================
// Kernel 2: zero-init fanout region of d_out
// =====================================================================
__global__ void zero_kernel(float* __restrict__ p, int n) {
  int i = blockIdx.x * blockDim.x + threadIdx.x;
  if (i < n) p[i] = 0.f;
}

// =====================================================================
// Kernel 3: per-expert exact radix select of k-th largest key
// =====================================================================
__global__ __launch_bounds__(1024)
void radix_select_kernel(const float* __restrict__ logitsT,
                         unsigned* __restrict__ thrbuf) {
  const int e = blockIdx.x;
  __shared__ unsigned hist[2048];
  __shared__ unsigned s_prefix, s_remaining, s_above;
  const float* col = logitsT + (size_t)e * NTOK;
  if (threadIdx.x == 0) { s_prefix = 0u; s_remaining = KSEL; s_above = 0u; }
  __syncthreads();

  for (int pass = 0; pass < 3; ++pass) {
    for (int i = threadIdx.x; i < 2048; i += 1024) hist[i] = 0u;
    __syncthreads();
    const unsigned prefix = s_prefix;
    for (int t = threadIdx.x; t < NTOK; t += 1024) {
      unsigned u = fkey(col[t]);
      unsigned bin; bool ok;
      if (pass == 0)      { bin = u >> 21;            ok = true; }
      else if (pass == 1) { bin = (u >> 10) & 0x7FFu; ok = ((u >> 21) == prefix); }
      else                { bin = u & 0x3FFu;         ok = ((u >> 10) == prefix); }
      if (ok) atomicAdd(&hist[bin], 1u);
    }
    __syncthreads();
    if (threadIdx.x == 0) {
      const int nb = (pass == 2) ? 1024 : 2048;
      unsigned cum = 0u; int b = nb - 1;
      for (; b > 0; --b) {
        unsigned c = hist[b];
        if (cum + c >= s_remaining) break;
        cum += c;
      }
      s_above += cum;
      s_remaining -= cum;
      s_prefix = (pass == 0) ? (unsigned)b
               : (pass == 1) ? ((s_prefix << 11) | (unsigned)b)
                             : ((s_prefix << 10) | (unsigned)b);
    }
    __syncthreads();
  }
  if (threadIdx.x == 0) { thrbuf[2 * e] = s_prefix; thrbuf[2 * e + 1] = s_above; }
}

// ---------------------------------------------------------------------
// bitonic sort, 4096 u64 descending, 1024 threads
// ---------------------------------------------------------------------
__device__ void bitonic4096_desc(unsigned long long* buf) {
  const int n = 4096;
  for (int size = 2; size <= n; size <<= 1) {
    for (int stride = size >> 1; stride > 0; stride >>= 1) {
      __syncthreads();
      for (int i = threadIdx.x; i < n; i += 1024) {
        int j = i ^ stride;
        if (j > i) {
          bool desc = ((i & size) == 0);
          unsigned long long a = buf[i], b = buf[j];
          if ((a < b) == desc) { buf[i] = b; buf[j] = a; }
        }
      }
    }
  }
  __syncthreads();
}

// =====================================================================
// Kernel 4: per-expert compaction + top_k ordering (value desc, idx asc)
// =====================================================================
__global__ __launch_bounds__(1024)
void select_compact_kernel(const float* __restrict__ logitsT,
                           const unsigned* __restrict__ thrbuf,
                           int* __restrict__ topk,
                           int* __restrict__ idx_out,
                           float* __restrict__ w_out,
                           float* __restrict__ fanout) {
  const int e = blockIdx.x;
  extern __shared__ char smemraw[];
  unsigned long long* strictBuf = (unsigned long long*)smemraw; // 4096
  unsigned long long* eqBuf     = strictBuf + KSEL;             // 4096
  __shared__ int s_ns, s_ne;
  if (threadIdx.x == 0) { s_ns = 0; s_ne = 0; }
  __syncthreads();

  const unsigned thr = thrbuf[2 * e];
  const float* col = logitsT + (size_t)e * NTOK;
  for (int t = threadIdx.x; t < NTOK; t += 1024) {
    unsigned u = fkey(col[t]);
    if (u > thr) {
      int p = atomicAdd(&s_ns, 1);
      if (p < KSEL)
        strictBuf[p] = ((unsigned long long)u << 32) | (unsigned long long)(0xFFFFFFFFu - (unsigned)t);
    } else if (u == thr) {
      int q = atomicAdd(&s_ne, 1);
      if (q < KSEL)
        eqBuf[q] = ((unsigned long long)u << 32) | (unsigned long long)(0xFFFFFFFFu - (unsigned)t);
    }
  }
  __syncthreads();
  const int ns = min(s_ns, KSEL), ne = min(s_ne, KSEL);
  for (int i = ns + threadIdx.x; i < KSEL; i += 1024) strictBuf[i] = 0ull;
  for (int i = ne + threadIdx.x; i < KSEL; i += 1024) eqBuf[i] = 0ull;
  __syncthreads();

  bitonic4096_desc(strictBuf);
  bitonic4096_desc(eqBuf);

  for (int j = threadIdx.x; j < KSEL; j += 1024) {
    unsigned long long v = (j < ns) ? strictBuf[j] : eqBuf[j - ns];
    int idx = (int)(0xFFFFFFFFu - (unsigned)(v & 0xFFFFFFFFull));
    float f = col[idx];
    float w = 1.0f / (1.0f + expf(-f));
    topk[(size_t)e * KSEL + j]    = idx;
    idx_out[(size_t)e * KSEL + j] = idx;   // int32 bit pattern in output
    w_out[(size_t)e * KSEL + j]   = w;
    atomicAdd(&fanout[idx], 1.0f);
  }
}

// =====================================================================
// Kernel 5: grouped expert FFN, fp32 WMMA + TDM async staging.
//   grid = (64 tiles of 64 tokens, 64 experts); block = 128 threads.
//   Phase A: TDM gather-mode load (16 token rows / wave, 16-bit indices)
//   Phase B: H = relu(X * W1^T), W1 slices via double-buffered TDM loads
//   Phase C: Y = H * W2^T,       W2 slices via double-buffered TDM loads
// =====================================================================
__global__ __launch_bounds__(128)
void ffn_kernel(const float* __restrict__ x, const float* __restrict__ w1,
                const float* __restrict__ w2, const int* __restrict__ topk,
                float* __restrict__ hout) {
  extern __shared__ char smemraw[];
  float* Xs = (float*)smemraw;                 // 64 x XS_STRIDE
  float* Hs = Xs + 64 * XS_STRIDE;             // 64 x HS_STRIDE
  float* Wb = Hs + 64 * HS_STRIDE;             // 2 x 16 x WS_STRIDE (ping-pong)
  const int tid  = threadIdx.x;
  const int e    = blockIdx.y;
  const int row0 = blockIdx.x * 64;
  const int wave = tid >> 5, lane = tid & 31;
  const int ml = lane & 15;
  const int ko = (lane >> 4) << 1;
  const int m0 = wave * 16;

  // ---- Phase A: token gather via TDM gather mode (one D# per wave) ----
  {
    const int* tk = topk + (size_t)e * KSEL + row0 + wave * 16;
    i32x4 g2, g3;
    #pragma unroll
    for (int j = 0; j < 4; ++j) {
      unsigned a0 = (unsigned)tk[2 * j]     & 0xFFFFu;
      unsigned a1 = (unsigned)tk[2 * j + 1] & 0xFFFFu;
      g2[j] = (int)(a0 | (a1 << 16));          // row_index_0..7
      unsigned b0 = (unsigned)tk[8 + 2 * j]     & 0xFFFFu;
      unsigned b1 = (unsigned)tk[8 + 2 * j + 1] & 0xFFFFu;
      g3[j] = (int)(b0 | (b1 << 16));          // row_index_8..15
    }
    unsigned lds_off = (unsigned)(uintptr_t)(Xs + wave * 16 * XS_STRIDE);
    u32x4 g0 = tdm_g0(lds_off, x, /*gather16=*/true);
    // rows of 256 f32, tensor height NTOK, 16 indices, pad -> stride 257
    i32x8 g1 = tdm_g1(/*dsz=*/2, /*ivl=*/7, /*amt=*/0,
                      CDIM, NTOK, CDIM, /*nidx=*/16, CDIM);
    tdm_load(g0, g1, g2, g3);
    __builtin_amdgcn_s_wait_tensorcnt(0);
  }
  __syncthreads();

  const i32x4 z4 = {0, 0, 0, 0};

  // ---- Phase B: GEMM1 + ReLU -> Hs (double-buffered W1 slices) ----
  const float* w1e = w1 + (size_t)e * DDIM * CDIM;
  if (wave == 0) {
    u32x4 g0 = tdm_g0((unsigned)(uintptr_t)Wb, w1e, false);
    i32x8 g1 = tdm_g1(2, 7, 0, CDIM, 16, CDIM, 16, CDIM);
    tdm_load(g0, g1, z4, z4);
    __builtin_amdgcn_s_wait_tensorcnt(0);
  }
  __syncthreads();

  for (int nt = 0; nt < DDIM / 16; ++nt) {
    const int cur = nt & 1;
    if (wave == 0 && nt + 1 < DDIM / 16) {    // prefetch next slice via TDM
      u32x4 g0 = tdm_g0((unsigned)(uintptr_t)(Wb + (1 - cur) * 16 * WS_STRIDE),
                        w1e + (size_t)(nt + 1) * 16 * CDIM, false);
      i32x8 g1 = tdm_g1(2, 7, 0, CDIM, 16, CDIM, 16, CDIM);
      tdm_load(g0, g1, z4, z4);
    }

    v8f acc = {0.f, 0.f, 0.f, 0.f, 0.f, 0.f, 0.f, 0.f};
    const float* arow = Xs + (m0 + ml) * XS_STRIDE;
    const float* brow = Wb + cur * 16 * WS_STRIDE + ml * WS_STRIDE;
    #pragma unroll 8
    for (int k0 = 0; k0 < CDIM; k0 += 4) {
      v2f a; a.x = arow[k0 + ko]; a.y = arow[k0 + ko + 1];
      v2f b; b.x = brow[k0 + ko]; b.y = brow[k0 + ko + 1];
      acc = wmma4(a, b, acc);
    }
    #pragma unroll
    for (int r = 0; r < 8; ++r) {
      int m = m0 + r + ((lane >> 4) << 3);
      Hs[m * HS_STRIDE + nt * 16 + ml] = fmaxf(acc[r], 0.f);
    }
    if (wave == 0) __builtin_amdgcn_s_wait_tensorcnt(0);
    __syncthreads();
  }

  // ---- Phase C: GEMM2 -> global (double-buffered W2 slices) ----
  const float* w2e = w2 + (size_t)e * CDIM * DDIM;
  if (wave == 0) {
    u32x4 g0 = tdm_g0((unsigned)(uintptr_t)Wb, w2e, false);
    // rows of 128 f32, pad 4 dwords each 128 -> LDS stride 132
    i32x8 g1 = tdm_g1(2, 6, 3, DDIM, 16, DDIM, 16, DDIM);
    tdm_load(g0, g1, z4, z4);
    __builtin_amdgcn_s_wait_tensorcnt(0);
  }
  __syncthreads();

  for (int nt = 0; nt < CDIM / 16; ++nt) {
    const int cur = nt & 1;
    if (wave == 0 && nt + 1 < CDIM / 16) {
      u32x4 g0 = tdm_g0((unsigned)(uintptr_t)(Wb + (1 - cur) * 16 * WS_STRIDE),
                        w2e + (size_t)(nt + 1) * 16 * DDIM, false);
      i32x8 g1 = tdm_g1(2, 6, 3, DDIM, 16, DDIM, 16, DDIM);
      tdm_load(g0, g1, z4, z4);
    }

    v8f acc = {0.f, 0.f, 0.f, 0.f, 0.f, 0.f, 0.f, 0.f};
    const float* arow = Hs + (m0 + ml) * HS_STRIDE;
    const float* brow = Wb + cur * 16 * WS_STRIDE + ml * HS_STRIDE;
    #pragma unroll 8
    for (int k0 = 0; k0 < DDIM; k0 += 4) {
      v2f a; a.x = arow[k0 + ko]; a.y = arow[k0 + ko + 1];
      v2f b; b.x = brow[k0 + ko]; b.y = brow[k0 + ko + 1];
      acc = wmma4(a, b, acc);
    }
    #pragma unroll
    for (int r = 0; r < 8; ++r) {
      int m = m0 + r + ((lane >> 4) << 3);
      size_t orow = (size_t)e * KSEL + row0 + m;
      hout[orow * CDIM + nt * 16 + ml] = acc[r];
    }
    if (wave == 0) __builtin_amdgcn_s_wait_tensorcnt(0);
    __syncthreads();
  }
}

// =====================================================================
// launch
// =====================================================================
extern "C" void kernel_launch(void* const* d_in, const int* in_sizes, int n_in,
                              void* d_out, int out_size, void* d_ws, size_t ws_size,
                              hipStream_t stream) {
  const float* x  = (const float*)d_in[0];   // (B,T,C)
  const float* rw = (const float*)d_in[1];   // (E,C)
  const float* w1 = (const float*)d_in[2];   // (E,D,C)
  const float* w2 = (const float*)d_in[3];   // (E,C,D)

  // d_out = h_flat (E*K*C) | indices (E*K, int32 bits) | weights (E*K) | fanout (N)
  float* hout    = (float*)d_out;
  int*   idx_out = (int*)(hout + (size_t)NEXP * KSEL * CDIM);
  float* w_out   = (float*)d_out + (size_t)NEXP * KSEL * CDIM + (size_t)NEXP * KSEL;
  float* fanout  = w_out + (size_t)NEXP * KSEL;

  // workspace: logitsT (E x N f32) | thrbuf (2E u32) | topk (E x K i32)
  float*    logitsT = (float*)d_ws;
  unsigned* thrbuf  = (unsigned*)(logitsT + (size_t)NEXP * NTOK);
  int*      topk    = (int*)(thrbuf + 2 * NEXP);

  const size_t router_lds = (size_t)(64 * XS_STRIDE + 16 * WS_STRIDE) * sizeof(float);
  const size_t ffn_lds    = (size_t)(64 * XS_STRIDE + 64 * HS_STRIDE +
                                     2 * 16 * WS_STRIDE) * sizeof(float);
  const size_t sel_lds    = (size_t)2 * KSEL * sizeof(unsigned long long);

  router_kernel<<<NTOK / 64, 128, router_lds, stream>>>(x, rw, logitsT);
  zero_kernel<<<NTOK / 256, 256, 0, stream>>>(fanout, NTOK);
  radix_select_kernel<<<NEXP, 1024, 0, stream>>>(logitsT, thrbuf);
  select_compact_kernel<<<NEXP, 1024, sel_lds, stream>>>(logitsT, thrbuf, topk,
                                                         idx_out, w_out, fanout);
  ffn_kernel<<<dim3(64, NEXP), 128, ffn_lds, stream>>>(x, w1, w2, topk, hout);
}